// GCN_53893249630533
// MI455X (gfx1250) — compile-verified
//
#include <hip/hip_runtime.h>
#include <hip/hip_bf16.h>

// ---------------------------------------------------------------------------
// ChebNet GCN forward for MI455X (gfx1250, wave32, WMMA).
//   - dense N x K @ K x 256 GEMMs on V_WMMA_F32_16X16X4_F32 (exact fp32),
//     A and B staged in LDS, 16(M)x32(N) tile per wave (A-fragment reuse)
//   - sparse propagation via coalesced gather + f32 atomic scatter
//   - tiny MLP head in scalar code
// ---------------------------------------------------------------------------

typedef float v2f __attribute__((ext_vector_type(2)));
typedef float v8f __attribute__((ext_vector_type(8)));

#define G_ACCUM 1
#define G_BIAS  2
#define G_RELU  4

#define BLOCK_M 128          // 8 waves * 16 rows
#define BLOCK_N 32           // 2 x 16-col tiles per wave
#define KC      32           // K-chunk staged in LDS
#define SAS     36           // padded LDS row stride (words): even, 36%4==0,
                             // 16/32 consecutive rows map to distinct banks

// ---------------- dense GEMM:  C[M,N] (+)= A[M,K] @ W[K,N] (+bias)(relu) ----
template <int FLAGS>
__global__ __launch_bounds__(256)
void gemm_f32_wmma(const float* __restrict__ A, const float* __restrict__ W,
                   float* __restrict__ C, const float* __restrict__ bias,
                   int M, int K, int N) {
  __shared__ float sA[BLOCK_M * SAS];   // A tile, row-major [mrow][k]
  __shared__ float sB[BLOCK_N * SAS];   // B tile, TRANSPOSED [col][k]

  const int tid  = threadIdx.x;
  const int wave = tid >> 5;
  const int lane = tid & 31;
  const int l16  = lane & 15;
  const int hi   = lane >> 4;            // 0: K pair {0,1}, 1: K pair {2,3}

  const int m_blk = blockIdx.x * BLOCK_M;
  const int n_blk = blockIdx.y * BLOCK_N;
  const int m0    = m_blk + wave * 16;   // wave-uniform
  const bool live = (m0 < M);            // M multiple of 16 -> whole tile valid

  v8f acc0 = {};
  v8f acc1 = {};

  for (int k0 = 0; k0 < K; k0 += KC) {
    __syncthreads();
    // A[m_blk..+127][k0..k0+31] -> LDS (float4, coalesced along K)
    for (int i = tid; i < BLOCK_M * (KC / 4); i += 256) {
      const int r  = i >> 3;             // row in tile
      const int c4 = (i & 7) << 2;       // col*4 in chunk
      const int gr = m_blk + r;
      float x0 = 0.f, x1 = 0.f, x2 = 0.f, x3 = 0.f;
      if (gr < M) {
        const float4 v = *reinterpret_cast<const float4*>(A + (size_t)gr * K + k0 + c4);
        x0 = v.x; x1 = v.y; x2 = v.z; x3 = v.w;
      }
      float* p = &sA[r * SAS + c4];
      p[0] = x0; p[1] = x1; p[2] = x2; p[3] = x3;
    }
    // W[k0..k0+31][n_blk..+31] -> LDS transposed (coalesced along N)
    for (int i = tid; i < KC * BLOCK_N; i += 256) {
      const int k = i >> 5;              // /BLOCK_N
      const int n = i & 31;
      sB[n * SAS + k] = W[(size_t)(k0 + k) * N + n_blk + n];
    }
    __syncthreads();

    if (live) {
      const float* sArow = &sA[(wave * 16 + l16) * SAS];
      const float* sBc0  = &sB[l16 * SAS];          // n-tile 0: col = l16
      const float* sBc1  = &sB[(16 + l16) * SAS];   // n-tile 1: col = 16+l16
#pragma unroll
      for (int kk = 0; kk < KC; kk += 4) {
        const int kb = kk + hi * 2;
        // A fragment (16x4 f32): lanes 0-15 K={0,1}, lanes 16-31 K={2,3}
        const float2 av = *reinterpret_cast<const float2*>(sArow + kb);
        v2f a; a.x = av.x; a.y = av.y;
        // B fragments (4x16 f32): row k striped over lanes in N, same K split
        const float2 b0v = *reinterpret_cast<const float2*>(sBc0 + kb);
        const float2 b1v = *reinterpret_cast<const float2*>(sBc1 + kb);
        v2f b0; b0.x = b0v.x; b0.y = b0v.y;
        v2f b1; b1.x = b1v.x; b1.y = b1v.y;
        acc0 = __builtin_amdgcn_wmma_f32_16x16x4_f32(false, a, false, b0,
                                                     (short)0, acc0, false, false);
        acc1 = __builtin_amdgcn_wmma_f32_16x16x4_f32(false, a, false, b1,
                                                     (short)0, acc1, false, false);
      }
    }
  }

  if (!live) return;

  const int col0 = n_blk + l16;
  const int col1 = col0 + 16;
  const float bv0 = (FLAGS & G_BIAS) ? bias[col0] : 0.f;
  const float bv1 = (FLAGS & G_BIAS) ? bias[col1] : 0.f;
#pragma unroll
  for (int r = 0; r < 8; ++r) {
    const int row = m0 + r + hi * 8;     // C/D layout: lanes 0-15 M=r, 16-31 M=r+8
    const size_t i0 = (size_t)row * N + col0;
    const size_t i1 = (size_t)row * N + col1;
    float v0 = acc0[r] + bv0;
    float v1 = acc1[r] + bv1;
    if (FLAGS & G_ACCUM) { v0 += C[i0]; v1 += C[i1]; }
    if (FLAGS & G_RELU)  { v0 = fmaxf(v0, 0.f); v1 = fmaxf(v1, 0.f); }
    C[i0] = v0;
    C[i1] = v1;
  }
}

// ---------------- graph normalization --------------------------------------
__global__ void deg_kernel(const int* __restrict__ src, float* __restrict__ deg, int E) {
  int e = blockIdx.x * blockDim.x + threadIdx.x;
  if (e < E) atomicAdd(&deg[src[e]], 1.0f);
}

__global__ void dis_kernel(const float* __restrict__ deg, float* __restrict__ dis, int N) {
  int i = blockIdx.x * blockDim.x + threadIdx.x;
  if (i < N) {
    float d = deg[i];
    dis[i] = (d > 0.f) ? rsqrtf(fmaxf(d, 1.f)) : 0.f;
  }
}

__global__ void ew_kernel(const int* __restrict__ src, const int* __restrict__ dst,
                          const float* __restrict__ dis, float* __restrict__ w, int E) {
  int e = blockIdx.x * blockDim.x + threadIdx.x;
  if (e < E) w[e] = -dis[src[e]] * dis[dst[e]];
}

// ---------------- prop: out[dst] += w * h[src]  (out pre-zeroed) ------------
__global__ void spmm_kernel(const float* __restrict__ h, const int* __restrict__ src,
                            const int* __restrict__ dst, const float* __restrict__ w,
                            float* __restrict__ out, int E, int H) {
  const long gid = (long)blockIdx.x * blockDim.x + threadIdx.x;
  const int cpe = H >> 2;
  const long total = (long)E * cpe;
  if (gid >= total) return;
  const int e = (int)(gid / cpe);
  const int c = ((int)(gid % cpe)) << 2;
  const int s = src[e], d = dst[e];
  const float ww = w[e];
  const float4 v = *reinterpret_cast<const float4*>(h + (size_t)s * H + c);
  float* op = out + (size_t)d * H + c;
  atomicAdd(op + 0, v.x * ww);
  atomicAdd(op + 1, v.y * ww);
  atomicAdd(op + 2, v.z * ww);
  atomicAdd(op + 3, v.w * ww);
}

// ---------------- Chebyshev recurrence elementwise: t = 2*t - prev ----------
__global__ void cheb_comb(float* __restrict__ t, const float* __restrict__ prev, long n) {
  long i = (long)blockIdx.x * blockDim.x + threadIdx.x;
  if (i < n) t[i] = 2.f * t[i] - prev[i];
}

// ---------------- global mean pool -----------------------------------------
__global__ void pool_sum(const float* __restrict__ h, const int* __restrict__ batch,
                         float* __restrict__ g, int N, int H) {
  long gid = (long)blockIdx.x * blockDim.x + threadIdx.x;
  if (gid >= (long)N * H) return;
  int i = (int)(gid / H), f = (int)(gid % H);
  atomicAdd(&g[(size_t)batch[i] * H + f], h[gid]);
}

__global__ void pool_cnt(const int* __restrict__ batch, float* __restrict__ cnt, int N) {
  int i = blockIdx.x * blockDim.x + threadIdx.x;
  if (i < N) atomicAdd(&cnt[batch[i]], 1.f);
}

__global__ void pool_div(float* __restrict__ g, const float* __restrict__ cnt, int G, int H) {
  int gid = blockIdx.x * blockDim.x + threadIdx.x;
  if (gid < G * H) g[gid] /= fmaxf(cnt[gid / H], 1.f);
}

// ---------------- small MLP head -------------------------------------------
__global__ void fc_relu(const float* __restrict__ in, const float* __restrict__ W,
                        const float* __restrict__ b, float* __restrict__ out,
                        int rows, int K, int N, int relu) {
  int gid = blockIdx.x * blockDim.x + threadIdx.x;
  if (gid >= rows * N) return;
  int r = gid / N, c = gid % N;
  float s = b[c];
  for (int k = 0; k < K; ++k) s += in[(size_t)r * K + k] * W[(size_t)k * N + c];
  out[gid] = relu ? fmaxf(s, 0.f) : s;
}

__global__ void head_logsoftmax(const float* __restrict__ in, const float* __restrict__ W,
                                const float* __restrict__ b, float* __restrict__ out,
                                int rows, int K, int C) {
  int r = blockIdx.x * blockDim.x + threadIdx.x;
  if (r >= rows) return;
  float lg[16];
  for (int c = 0; c < C; ++c) {
    float s = b[c];
    for (int k = 0; k < K; ++k) s += in[(size_t)r * K + k] * W[(size_t)k * C + c];
    lg[c] = s;
  }
  float mx = lg[0];
  for (int c = 1; c < C; ++c) mx = fmaxf(mx, lg[c]);
  float se = 0.f;
  for (int c = 0; c < C; ++c) se += __expf(lg[c] - mx);
  float lse = mx + __logf(se);
  for (int c = 0; c < C; ++c) out[(size_t)r * C + c] = lg[c] - lse;
}

// ---------------- host-side orchestration ----------------------------------
static inline void launch_gemm(const float* A, const float* W, float* C, const float* bias,
                               int M, int K, int N, int flags, hipStream_t s) {
  dim3 grid((M + BLOCK_M - 1) / BLOCK_M, N / BLOCK_N);
  switch (flags) {
    case 0:
      gemm_f32_wmma<0><<<grid, 256, 0, s>>>(A, W, C, bias, M, K, N); break;
    case G_ACCUM:
      gemm_f32_wmma<G_ACCUM><<<grid, 256, 0, s>>>(A, W, C, bias, M, K, N); break;
    case G_BIAS | G_RELU:
      gemm_f32_wmma<G_BIAS | G_RELU><<<grid, 256, 0, s>>>(A, W, C, bias, M, K, N); break;
    default:
      gemm_f32_wmma<G_ACCUM | G_BIAS | G_RELU><<<grid, 256, 0, s>>>(A, W, C, bias, M, K, N);
      break;
  }
}

static inline void launch_prop(const float* h, int H, float* out,
                               const int* src, const int* dst, const float* w,
                               int N, int E, hipStream_t s) {
  hipMemsetAsync(out, 0, (size_t)N * H * sizeof(float), s);
  long total = (long)E * (H / 4);
  int blocks = (int)((total + 255) / 256);
  spmm_kernel<<<blocks, 256, 0, s>>>(h, src, dst, w, out, E, H);
}

static void run_cheb(const float* in, int indim, const float* Wk, int Korder,
                     const float* bias, float* out,
                     const int* src, const int* dst, const float* w,
                     float* bufA, float* bufB, float* bufC,
                     int N, int E, hipStream_t s) {
  const int Nout = 256;
  const size_t wstride = (size_t)indim * Nout;

  // out = T0 @ W[0]           (T0 = in)
  launch_gemm(in, Wk, out, nullptr, N, indim, Nout, 0, s);

  // T1 = L_hat @ in ; out += T1 @ W[1]
  launch_prop(in, indim, bufB, src, dst, w, N, E, s);
  int fl = G_ACCUM | ((Korder == 2) ? (G_BIAS | G_RELU) : 0);
  launch_gemm(bufB, Wk + wstride, out, bias, N, indim, Nout, fl, s);

  const float* prev = in;
  const float* cur  = bufB;
  float* pool3[3] = {bufC, bufA, bufB};   // rotation never clobbers prev/cur
  for (int k = 2; k < Korder; ++k) {
    float* tmp = pool3[(k - 2) % 3];
    launch_prop(cur, indim, tmp, src, dst, w, N, E, s);        // tmp = L_hat @ Tk-1
    long n = (long)N * indim;
    cheb_comb<<<(int)((n + 255) / 256), 256, 0, s>>>(tmp, prev, n);  // tmp = 2*tmp - Tk-2
    int fl2 = G_ACCUM | ((k == Korder - 1) ? (G_BIAS | G_RELU) : 0);
    launch_gemm(tmp, Wk + (size_t)k * wstride, out, bias, N, indim, Nout, fl2, s);
    prev = cur; cur = tmp;
  }
}

extern "C" void kernel_launch(void* const* d_in, const int* in_sizes, int n_in,
                              void* d_out, int out_size, void* d_ws, size_t ws_size,
                              hipStream_t stream) {
  // inputs (setup_inputs order)
  const float* x      = (const float*)d_in[0];
  const int*   eidx   = (const int*)  d_in[1];
  const int*   batch  = (const int*)  d_in[2];
  const float* W1     = (const float*)d_in[3];
  const float* b1     = (const float*)d_in[4];
  const float* W2     = (const float*)d_in[5];
  const float* b2     = (const float*)d_in[6];
  const float* W3     = (const float*)d_in[7];
  const float* b3     = (const float*)d_in[8];
  const float* fc1w   = (const float*)d_in[9];
  const float* fc1b   = (const float*)d_in[10];
  const float* fc2w   = (const float*)d_in[11];
  const float* fc2b   = (const float*)d_in[12];
  const float* fc3w   = (const float*)d_in[13];
  const float* fc3b   = (const float*)d_in[14];
  float* out = (float*)d_out;

  const int N = in_sizes[2];          // 50000
  const int E = in_sizes[1] / 2;      // 800000
  const int H = 256, G = 64, F = 128;
  const int* src = eidx;
  const int* dst = eidx + E;

  // workspace carve-up (floats)
  const size_t NH = (size_t)N * H;
  size_t need = (5 * NH + 2 * (size_t)N + (size_t)E +
                 (size_t)G * H + G + (size_t)G * 512 + (size_t)G * 256) * sizeof(float);
  if (ws_size < need) return;

  float* wsf  = (float*)d_ws;
  float* bufA = wsf;            // cheb temp
  float* bufB = bufA + NH;      // cheb temp
  float* bufC = bufB + NH;      // cheb temp
  float* H1   = bufC + NH;      // layer activations
  float* H2   = H1   + NH;
  float* deg  = H2   + NH;
  float* dis  = deg  + N;
  float* w    = dis  + N;
  float* g    = w    + E;
  float* cnt  = g    + (size_t)G * H;
  float* f1   = cnt  + G;
  float* f2   = f1   + (size_t)G * 512;

  // ---- normalized edge weights  w = -d^-1/2[src] * d^-1/2[dst]
  hipMemsetAsync(deg, 0, N * sizeof(float), stream);
  deg_kernel<<<(E + 255) / 256, 256, 0, stream>>>(src, deg, E);
  dis_kernel<<<(N + 255) / 256, 256, 0, stream>>>(deg, dis, N);
  ew_kernel<<<(E + 255) / 256, 256, 0, stream>>>(src, dst, dis, w, E);

  // ---- three ChebConv layers (K = 3, 4, 5), ReLU fused into final GEMM
  run_cheb(x,  F, W1, 3, b1, H1, src, dst, w, bufA, bufB, bufC, N, E, stream);
  run_cheb(H1, H, W2, 4, b2, H2, src, dst, w, bufA, bufB, bufC, N, E, stream);
  run_cheb(H2, H, W3, 5, b3, H1, src, dst, w, bufA, bufB, bufC, N, E, stream);

  // ---- global mean pool over graphs
  hipMemsetAsync(g, 0, (size_t)G * H * sizeof(float), stream);
  hipMemsetAsync(cnt, 0, G * sizeof(float), stream);
  {
    long n = (long)N * H;
    pool_sum<<<(int)((n + 255) / 256), 256, 0, stream>>>(H1, batch, g, N, H);
    pool_cnt<<<(N + 255) / 256, 256, 0, stream>>>(batch, cnt, N);
    pool_div<<<(G * H + 255) / 256, 256, 0, stream>>>(g, cnt, G, H);
  }

  // ---- MLP head + log_softmax
  fc_relu<<<(G * 512 + 255) / 256, 256, 0, stream>>>(g,  fc1w, fc1b, f1, G, 256, 512, 1);
  fc_relu<<<(G * 256 + 255) / 256, 256, 0, stream>>>(f1, fc2w, fc2b, f2, G, 512, 256, 1);
  head_logsoftmax<<<1, G, 0, stream>>>(f2, fc3w, fc3b, out, G, 256, 10);
}